// FIRSTLoss_56418690400290
// MI455X (gfx1250) — compile-verified
//
#include <hip/hip_runtime.h>
#include <hip/hip_bf16.h>
#include <stdint.h>

#define IGNORE_INDEX (-100)

// ---------------------------------------------------------------------------
// CDNA5 async global->LDS DMA helpers (gfx1250: ASYNCcnt-tracked copies).
// Each lane moves 16B (b128) or 4B (b32) from its global address to its LDS
// byte address. We wait with s_wait_asynccnt 0 then barrier for cross-wave
// visibility.
// ---------------------------------------------------------------------------
__device__ __forceinline__ void async_g2lds_b128(uint32_t lds_off, const void* gaddr) {
#if defined(__gfx1250__)
    asm volatile("global_load_async_to_lds_b128 %0, %1, off"
                 :: "v"(lds_off), "v"(gaddr) : "memory");
#else
    (void)lds_off; (void)gaddr;
#endif
}

__device__ __forceinline__ void async_g2lds_b32(uint32_t lds_off, const void* gaddr) {
#if defined(__gfx1250__)
    asm volatile("global_load_async_to_lds_b32 %0, %1, off"
                 :: "v"(lds_off), "v"(gaddr) : "memory");
#else
    (void)lds_off; (void)gaddr;
#endif
}

__device__ __forceinline__ void wait_async_zero() {
#if defined(__gfx1250__)
    asm volatile("s_wait_asynccnt 0" ::: "memory");
#endif
}

// wave32 butterfly reductions
__device__ __forceinline__ float wave_max(float v) {
    for (int off = 16; off > 0; off >>= 1) v = fmaxf(v, __shfl_xor(v, off, 32));
    return v;
}
__device__ __forceinline__ float wave_sum(float v) {
    for (int off = 16; off > 0; off >>= 1) v += __shfl_xor(v, off, 32);
    return v;
}

// ---------------------------------------------------------------------------
// Kernel 1 (LDS path): one workgroup per row of [rows, V].
//   stage row into LDS via async DMA -> max pass -> sum-exp pass -> nll.
// ---------------------------------------------------------------------------
__global__ void ce_row_lds(const float* __restrict__ logits,
                           const int*   __restrict__ labels,
                           int V,
                           float* __restrict__ nll_out,
                           int*   __restrict__ valid_out) {
    extern __shared__ __align__(16) unsigned char smem_raw[];
    float* rowbuf = reinterpret_cast<float*>(smem_raw);

    __shared__ float wred[16];
    __shared__ float bcast[2];

    const int row  = blockIdx.x;
    const int tid  = threadIdx.x;
    const int T    = blockDim.x;           // 512 -> 16 wave32s
    const int lane = tid & 31;
    const int wave = tid >> 5;

    const float* grow = logits + (size_t)row * (size_t)V;

    // LDS byte offset of rowbuf: low 32 bits of the generic pointer are the
    // workgroup-relative LDS address (aperture lives in the high 32 bits).
    const uint32_t lds_base = (uint32_t)(uintptr_t)(const void*)rowbuf;

    // ---- async DMA: global row -> LDS ----
    const int v4 = V >> 2;                 // 16B-aligned vec4 chunks
    for (int i = tid; i < v4; i += T) {
        async_g2lds_b128(lds_base + 16u * (uint32_t)i, (const void*)(grow + 4 * (size_t)i));
    }
    for (int i = (v4 << 2) + tid; i < V; i += T) {
        async_g2lds_b32(lds_base + 4u * (uint32_t)i, (const void*)(grow + i));
    }
    wait_async_zero();
    __syncthreads();

    // ---- pass 1: row max (from LDS) ----
    float m = -INFINITY;
    for (int i = tid; i < V; i += T) m = fmaxf(m, rowbuf[i]);
    m = wave_max(m);
    if (lane == 0) wred[wave] = m;
    __syncthreads();
    if (tid < 16) {                        // 16 waves when T == 512
        float t = wred[tid];
        for (int off = 8; off > 0; off >>= 1) t = fmaxf(t, __shfl_xor(t, off, 32));
        if (tid == 0) bcast[0] = t;
    }
    __syncthreads();
    const float rowmax = bcast[0];

    // ---- pass 2: sum of exp(x - max) (from LDS) ----
    float s = 0.0f;
    for (int i = tid; i < V; i += T) s += expf(rowbuf[i] - rowmax);
    s = wave_sum(s);
    __syncthreads();                       // wred reuse
    if (lane == 0) wred[wave] = s;
    __syncthreads();
    if (tid < 16) {
        float t = wred[tid];
        for (int off = 8; off > 0; off >>= 1) t += __shfl_xor(t, off, 32);
        if (tid == 0) bcast[1] = t;
    }
    __syncthreads();

    if (tid == 0) {
        const int lbl = labels[row];
        const int ok  = (lbl != IGNORE_INDEX) ? 1 : 0;
        const int sl  = (lbl >= 0 && lbl < V) ? lbl : 0;   // safe label
        const float xl = rowbuf[sl];
        nll_out[row]   = logf(bcast[1]) + rowmax - xl;     // logsumexp - x[label]
        valid_out[row] = ok;
    }
}

// ---------------------------------------------------------------------------
// Kernel 1 fallback (no LDS staging) for rows too large for LDS. Reads global
// twice (second pass mostly from L2). Not used at the reference sizes.
// ---------------------------------------------------------------------------
__global__ void ce_row_global(const float* __restrict__ logits,
                              const int*   __restrict__ labels,
                              int V,
                              float* __restrict__ nll_out,
                              int*   __restrict__ valid_out) {
    __shared__ float wred[16];
    __shared__ float bcast[2];

    const int row  = blockIdx.x;
    const int tid  = threadIdx.x;
    const int T    = blockDim.x;
    const int lane = tid & 31;
    const int wave = tid >> 5;

    const float* grow = logits + (size_t)row * (size_t)V;

    float m = -INFINITY;
    for (int i = tid; i < V; i += T) m = fmaxf(m, grow[i]);
    m = wave_max(m);
    if (lane == 0) wred[wave] = m;
    __syncthreads();
    if (tid < 16) {
        float t = wred[tid];
        for (int off = 8; off > 0; off >>= 1) t = fmaxf(t, __shfl_xor(t, off, 32));
        if (tid == 0) bcast[0] = t;
    }
    __syncthreads();
    const float rowmax = bcast[0];

    float s = 0.0f;
    for (int i = tid; i < V; i += T) s += expf(grow[i] - rowmax);
    s = wave_sum(s);
    __syncthreads();
    if (lane == 0) wred[wave] = s;
    __syncthreads();
    if (tid < 16) {
        float t = wred[tid];
        for (int off = 8; off > 0; off >>= 1) t += __shfl_xor(t, off, 32);
        if (tid == 0) bcast[1] = t;
    }
    __syncthreads();

    if (tid == 0) {
        const int lbl = labels[row];
        const int ok  = (lbl != IGNORE_INDEX) ? 1 : 0;
        const int sl  = (lbl >= 0 && lbl < V) ? lbl : 0;
        nll_out[row]   = logf(bcast[1]) + rowmax - grow[sl];
        valid_out[row] = ok;
    }
}

// ---------------------------------------------------------------------------
// Kernel 2: single block. Reduce per-row nll, compute RankNet pairwise loss
// (N^2 = 4096 scalar pairs), combine into the scalar output.
// ---------------------------------------------------------------------------
__global__ void finalize_kernel(const float* __restrict__ nll,
                                const int*   __restrict__ valid,
                                int rows,
                                const float* __restrict__ rlog,
                                const int*   __restrict__ ranks,
                                int N,
                                float* __restrict__ out) {
    const int tid  = threadIdx.x;
    const int T    = blockDim.x;           // 256 -> 8 wave32s
    const int lane = tid & 31;
    const int wave = tid >> 5;

    float s = 0.0f, c = 0.0f, rp = 0.0f;

    for (int i = tid; i < rows; i += T) {
        if (valid[i]) { s += nll[i]; c += 1.0f; }
    }

    for (int i = tid; i < N; i += T) {
        const float li = rlog[i];
        const float ri = (float)ranks[i];
        for (int j = i + 1; j < N; ++j) {
            const float rj = (float)ranks[j];
            if (ri < rj) {
                const float d  = rlog[j] - li;                       // logits[j] - logits[i]
                const float sp = fmaxf(d, 0.0f) + log1pf(expf(-fabsf(d))); // stable softplus
                rp += sp / (ri + rj);
            }
        }
    }

    s  = wave_sum(s);
    c  = wave_sum(c);
    rp = wave_sum(rp);

    __shared__ float rs[3][8];
    if (lane == 0) { rs[0][wave] = s; rs[1][wave] = c; rs[2][wave] = rp; }
    __syncthreads();

    if (tid == 0) {
        float S = 0.0f, C = 0.0f, R = 0.0f;
        const int nw = T >> 5;
        for (int w = 0; w < nw; ++w) { S += rs[0][w]; C += rs[1][w]; R += rs[2][w]; }
        const float lm_loss   = S / fmaxf(C, 1.0f);
        const float rank_loss = R / (float)N;
        out[0] = 1.0f * lm_loss + 10.0f * rank_loss;
    }
}

// ---------------------------------------------------------------------------
// Host launcher
// ---------------------------------------------------------------------------
extern "C" void kernel_launch(void* const* d_in, const int* in_sizes, int n_in,
                              void* d_out, int out_size, void* d_ws, size_t ws_size,
                              hipStream_t stream) {
    (void)n_in; (void)out_size; (void)ws_size;

    const float* lm_logits = (const float*)d_in[0];   // [rows, V] f32
    const int*   lm_labels = (const int*)  d_in[1];   // [rows] i32
    const float* rlog      = (const float*)d_in[2];   // [N] f32
    const int*   ranks     = (const int*)  d_in[3];   // [N] i32

    const int rows = in_sizes[1];                     // B*S
    const int V    = in_sizes[0] / rows;              // vocab
    const int N    = in_sizes[2];                     // ranking count

    float* nll_ws   = (float*)d_ws;                                     // rows floats
    int*   valid_ws = (int*)((char*)d_ws + (size_t)rows * sizeof(float)); // rows ints

    const size_t lds_bytes = (size_t)V * sizeof(float);
    if (lds_bytes <= 160u * 1024u) {
        ce_row_lds<<<rows, 512, lds_bytes, stream>>>(lm_logits, lm_labels, V,
                                                     nll_ws, valid_ws);
    } else {
        ce_row_global<<<rows, 512, 0, stream>>>(lm_logits, lm_labels, V,
                                                nll_ws, valid_ws);
    }

    finalize_kernel<<<1, 256, 0, stream>>>(nll_ws, valid_ws, rows,
                                           rlog, ranks, N, (float*)d_out);
}